// CapsuleLayer_25443386261974
// MI455X (gfx1250) — compile-verified
//
#include <hip/hip_runtime.h>
#include <hip/hip_bf16.h>

// ---------------------------------------------------------------------------
// CapsuleLayer on MI455X (gfx1250).
//
// The reference collapses exactly (routing is a no-op because delta is
// broadcast uniformly over the capsule axis, so softmax stays 1/16):
//   t[b,o,h,w] = (1/16) * ( conv3x3(x, sum_c W[c,o]) + sum_c b[c,o] )
//   out        = t*|t| / (1 + t*t)
// Single 64ch->16ch 3x3 VALID conv over [64,64,66,66] -> [64,16,64,64],
// implemented as implicit GEMM with V_WMMA_F32_16X16X4_F32 (full f32).
// ---------------------------------------------------------------------------

typedef float v2f __attribute__((ext_vector_type(2)));
typedef float v8f __attribute__((ext_vector_type(8)));

#define IN_CH     64
#define OUT_CH    16
#define HW_IN     66
#define HW_OUT    64
#define BATCH     64
#define KTOT      576              // 9 taps * 64 in channels
#define BMAT_ELEMS (KTOT * OUT_CH) // 9216 floats = 36 KB
#define X_IMG     (IN_CH * HW_IN * HW_IN)  // 278784 floats per batch image
#define X_PLANE   (HW_IN * HW_IN)          // 4356
#define WAVES_PER_WG 4

// --- prep: fold capsule sum into weights, reorder to K-major [k][o] --------
// K ordering: k = tap*64 + i  (tap = r*3+s, i = input channel)
__global__ __launch_bounds__(256) void caps_prep_kernel(
    const float* __restrict__ W,   // [16,16,64,3,3]
    const float* __restrict__ b,   // [16,16]
    float* __restrict__ Bmat,      // [576][16]
    float* __restrict__ bsum)      // [16]
{
    int t = blockIdx.x * blockDim.x + threadIdx.x;
    if (t < BMAT_ELEMS) {
        int o   = t & 15;
        int ki  = t >> 4;          // tap*64 + i
        int tap = ki >> 6;
        int i   = ki & 63;
        float s = 0.0f;
        #pragma unroll
        for (int c = 0; c < 16; ++c)
            s += W[((c * 16 + o) * 64 + i) * 9 + tap];
        Bmat[t] = s;
    }
    if (t < OUT_CH) {
        float s = 0.0f;
        #pragma unroll
        for (int c = 0; c < 16; ++c) s += b[c * 16 + t];
        bsum[t] = s;
    }
}

// --- main: implicit-GEMM conv via f32 WMMA ---------------------------------
// One wave -> one 16(pixels) x 16(channels) tile. 144 WMMA K=4 steps.
__global__ __launch_bounds__(WAVES_PER_WG * 32) void caps_conv_kernel(
    const float* __restrict__ x,     // [64,64,66,66]
    const float* __restrict__ Bmat,  // [576][16]
    const float* __restrict__ bsum,  // [16]
    float* __restrict__ out)         // [64,16,64,64]
{
    __shared__ float Bs[BMAT_ELEMS]; // 36 KB weights, shared by all waves

    const int tid = threadIdx.x;
    for (int idx = tid; idx < BMAT_ELEMS; idx += WAVES_PER_WG * 32)
        Bs[idx] = Bmat[idx];
    __syncthreads();

    const int lane = tid & 31;
    const int wave = tid >> 5;
    const int tile = blockIdx.x * WAVES_PER_WG + wave; // 16384 tiles total
    const int bIdx = tile >> 8;          // batch
    const int rem  = tile & 255;
    const int h    = rem >> 2;           // output row
    const int w0   = (rem & 3) << 4;     // output col base (16-wide segment)

    // ISA A-fragment layout (16x4 f32): M = lane%16; lanes<16 hold K pair
    // {0,1}, lanes>=16 hold {2,3} of each 4-chunk.
    const int msp   = lane & 15;         // spatial pixel within tile (M)
    const int khalf = (lane >> 4) << 1;  // 0 or 2
    const int ch    = lane & 15;         // output channel (N) for B/D frags

    const float* xb = x + (size_t)bIdx * X_IMG + (size_t)h * HW_IN + w0 + msp;

    v8f acc = {};
    #pragma unroll
    for (int tap = 0; tap < 9; ++tap) {
        const int r = tap / 3;
        const int s = tap - 3 * r;
        const float* xt = xb + r * HW_IN + s;           // x[b][.][h+r][w0+m+s]
        const float* bt = Bs + tap * (64 * OUT_CH);     // weights for this tap
        // prefetch next tap's first line into cache (global_prefetch_b8)
        if (tap < 8) __builtin_prefetch(xb + (r + ((s + 1) / 3)) * HW_IN + ((s + 1) % 3), 0, 1);
        #pragma unroll
        for (int step = 0; step < 16; ++step) {
            const int i0 = step * 4 + khalf;            // input channel pair
            v2f a, bf;
            a.x  = xt[(size_t)i0 * X_PLANE];
            a.y  = xt[(size_t)(i0 + 1) * X_PLANE];
            bf.x = bt[i0 * OUT_CH + ch];
            bf.y = bt[(i0 + 1) * OUT_CH + ch];
            acc = __builtin_amdgcn_wmma_f32_16x16x4_f32(
                /*neg_a=*/false, a, /*neg_b=*/false, bf,
                /*c_mod=*/(short)0, acc, /*reuse_a=*/false, /*reuse_b=*/false);
        }
    }

    // Epilogue: D layout -> lane holds channel `ch`, rows M = v + 8*(lane/16)
    // which are 8 consecutive output pixels -> two contiguous b128 stores.
    const int mrow0 = (lane >> 4) << 3;  // 0 or 8
    const float bb = bsum[ch];
    float res[8];
    #pragma unroll
    for (int v = 0; v < 8; ++v) {
        float t = (acc[v] + bb) * 0.0625f;       // 1/16 uniform routing weight
        res[v] = t * fabsf(t) / (1.0f + t * t);  // collapsed squash
    }
    float* op = out + (size_t)bIdx * (OUT_CH * HW_OUT * HW_OUT)
                    + (size_t)ch * (HW_OUT * HW_OUT)
                    + (size_t)h * HW_OUT + w0 + mrow0;
    *(float4*)(op)     = make_float4(res[0], res[1], res[2], res[3]);
    *(float4*)(op + 4) = make_float4(res[4], res[5], res[6], res[7]);
}

extern "C" void kernel_launch(void* const* d_in, const int* in_sizes, int n_in,
                              void* d_out, int out_size, void* d_ws, size_t ws_size,
                              hipStream_t stream) {
    const float* x = (const float*)d_in[0];   // [64,64,66,66]
    const float* W = (const float*)d_in[1];   // [16,16,64,3,3]
    const float* b = (const float*)d_in[2];   // [16,16]
    float* out  = (float*)d_out;              // [64*16*64*64]
    float* Bmat = (float*)d_ws;               // 9216 f32
    float* bsum = Bmat + BMAT_ELEMS;          // 16 f32

    caps_prep_kernel<<<(BMAT_ELEMS + 255) / 256, 256, 0, stream>>>(W, b, Bmat, bsum);

    // 64*64*4 = 16384 tiles, 4 waves/WG -> 4096 workgroups
    caps_conv_kernel<<<16384 / WAVES_PER_WG, WAVES_PER_WG * 32, 0, stream>>>(
        x, Bmat, bsum, out);
}